// CrossTransformerLayer_62139586839040
// MI455X (gfx1250) — compile-verified
//
#include <hip/hip_runtime.h>
#include <hip/hip_bf16.h>

typedef __attribute__((ext_vector_type(2))) float v2f;
typedef __attribute__((ext_vector_type(8))) float v8f;
typedef __attribute__((__vector_size__(16))) int v4i;

#define NF   128
#define EPS  1e-4f
// LDS row strides (in floats) chosen for bank-conflict-free access (64 banks):
#define KSTR 132   // S-loop reads stride KSTR per lane: banks 4*ln+4c+2hf all distinct
#define VSTR 144   // PV reads: 2*VSTR = 288 == 32 mod 64 -> lane halves disjoint
#define BSTR 144   // same pattern as V for GEMM B panels
#define PSTR 17    // A-layout P reads: 17*ln distinct mod 64 (gcd(17,64)=1)

// ---------------------------------------------------------------------------
// gfx1250 async global->LDS copy (ASYNCcnt path), with synchronous fallback.
// Probe-verified: first param is v4i in AS(1) ("__device__"); 4 args total.
// ---------------------------------------------------------------------------
#if defined(__has_builtin)
#if __has_builtin(__builtin_amdgcn_global_load_async_to_lds_b128)
#define HAVE_ASYNC_LDS 1
#endif
#endif

__device__ __forceinline__ void async_copy_b128(const float* gsrc, float* lds_dst) {
#ifdef HAVE_ASYNC_LDS
  const __attribute__((address_space(1))) float* g1 =
      (const __attribute__((address_space(1))) float*)gsrc;
  __attribute__((address_space(3))) float* l3 =
      (__attribute__((address_space(3))) float*)lds_dst;
  __builtin_amdgcn_global_load_async_to_lds_b128(
      (__attribute__((address_space(1))) v4i*)g1,
      (__attribute__((address_space(3))) v4i*)l3, 0, 0);
#else
  *(float4*)lds_dst = *(const float4*)gsrc;
#endif
}

__device__ __forceinline__ void async_wait0() {
#ifdef HAVE_ASYNC_LDS
#if defined(__has_builtin) && __has_builtin(__builtin_amdgcn_s_wait_asynccnt)
  __builtin_amdgcn_s_wait_asynccnt(0);
#else
  asm volatile("s_wait_asynccnt 0x0" ::: "memory");
#endif
#endif
}

// ---------------------------------------------------------------------------
// Kernel 1: submanifold 3x3x3 conv as gathered GEMM (K = 27*64 = 1728):
//   xdec[n,f] = sum_ko sum_c feat[nbr[n,ko], c] * Wp1[ko,c,f]
// 8 waves/block, wave owns 16 rows x 128 cols. Wp1[ko] slice async-staged.
// ---------------------------------------------------------------------------
__global__ void p1conv_wmma(const float* __restrict__ feat,  // [Nq,64]
                            const int*   __restrict__ nbr,   // [Nq,27]
                            const float* __restrict__ Wp1,   // [27,64,128]
                            float* __restrict__ xdec,        // [Nq,128]
                            int Nq) {
  __shared__ float Bs[64 * BSTR];  // 36 KB padded panel
  const int tid  = threadIdx.x;
  const int wave = tid >> 5;
  const int lane = tid & 31;
  const int ln   = lane & 15;
  const int hf   = lane >> 4;
  const int rowbase = blockIdx.x * 128 + wave * 16;

  v8f zero = {};
  v8f acc[8];
#pragma unroll
  for (int ft = 0; ft < 8; ++ft) acc[ft] = zero;

  for (int ko = 0; ko < 27; ++ko) {
    __syncthreads();  // previous panel fully consumed
    const float* src = Wp1 + (long)ko * 64 * NF;
    for (int g = tid; g < 64 * 32; g += 256) {   // 2048 B128 granules
      const int row = g >> 5, c4 = (g & 31) << 2;
      async_copy_b128(src + row * NF + c4, Bs + row * BSTR + c4);
    }
    async_wait0();
    __syncthreads();

    const int idx = nbr[(long)(rowbase + ln) * 27 + ko];
    const float* arow = feat + (long)idx * 64;

    for (int c0 = 0; c0 < 64; c0 += 4) {
      v2f a;
      a.x = arow[c0 + 2 * hf + 0];
      a.y = arow[c0 + 2 * hf + 1];
#pragma unroll
      for (int ft = 0; ft < 8; ++ft) {
        v2f bb;
        bb.x = Bs[(c0 + 2 * hf + 0) * BSTR + ft * 16 + ln];
        bb.y = Bs[(c0 + 2 * hf + 1) * BSTR + ft * 16 + ln];
        acc[ft] = __builtin_amdgcn_wmma_f32_16x16x4_f32(
            false, a, false, bb, (short)0, acc[ft], false, false);
      }
    }
  }

#pragma unroll
  for (int ft = 0; ft < 8; ++ft)
#pragma unroll
    for (int r = 0; r < 8; ++r)
      xdec[(long)(rowbase + r + 8 * hf) * NF + ft * 16 + ln] = acc[ft][r];
}

// ---------------------------------------------------------------------------
// Kernel 2: generic GEMM  C[M,128] = A[M,Kd] @ B[Kd,128]  (Kd in {64,128})
// ---------------------------------------------------------------------------
__global__ void gemm_wmma(const float* __restrict__ A,
                          const float* __restrict__ B,
                          float* __restrict__ C,
                          int M, int Kd) {
  __shared__ float Bs[64 * BSTR];
  const int tid  = threadIdx.x;
  const int wave = tid >> 5;
  const int lane = tid & 31;
  const int ln   = lane & 15;
  const int hf   = lane >> 4;
  const int rowbase = blockIdx.x * 128 + wave * 16;

  v8f zero = {};
  v8f acc[8];
#pragma unroll
  for (int ft = 0; ft < 8; ++ft) acc[ft] = zero;

  const float* arow = A + (long)(rowbase + ln) * Kd;

  for (int cbase = 0; cbase < Kd; cbase += 64) {
    __syncthreads();
    const float* src = B + (long)cbase * NF;
    for (int g = tid; g < 64 * 32; g += 256) {
      const int row = g >> 5, c4 = (g & 31) << 2;
      async_copy_b128(src + row * NF + c4, Bs + row * BSTR + c4);
    }
    async_wait0();
    __syncthreads();

    for (int c0 = 0; c0 < 64; c0 += 4) {
      v2f a;
      a.x = arow[cbase + c0 + 2 * hf + 0];
      a.y = arow[cbase + c0 + 2 * hf + 1];
#pragma unroll
      for (int ft = 0; ft < 8; ++ft) {
        v2f bb;
        bb.x = Bs[(c0 + 2 * hf + 0) * BSTR + ft * 16 + ln];
        bb.y = Bs[(c0 + 2 * hf + 1) * BSTR + ft * 16 + ln];
        acc[ft] = __builtin_amdgcn_wmma_f32_16x16x4_f32(
            false, a, false, bb, (short)0, acc[ft], false, false);
      }
    }
  }

#pragma unroll
  for (int ft = 0; ft < 8; ++ft)
#pragma unroll
    for (int r = 0; r < 8; ++r)
      C[(long)(rowbase + r + 8 * hf) * NF + ft * 16 + ln] = acc[ft][r];
}

// ---------------------------------------------------------------------------
// Kernel 3: single-pass flash attention, double-buffered async K/V staging.
// Block = 8 waves = 128 query rows of one batch; wave owns 16 rows.
// Q in registers; online softmax via 16-lane shfl butterflies; P converted
// C-layout -> A-layout through padded per-wave LDS scratch.
// ---------------------------------------------------------------------------
__global__ void attn_wmma(const float* __restrict__ Q,   // [B*nq, 128]
                          const float* __restrict__ Km,  // [B*nkv, 128]
                          const float* __restrict__ Vm,  // [B*nkv, 128]
                          float* __restrict__ O,         // [B*nq, 128]
                          int nq_per_b, int nkv_per_b) {
  __shared__ float Ks[2][16 * KSTR];   // 2 x 8.25 KB
  __shared__ float Vs[2][16 * VSTR];   // 2 x 9 KB
  __shared__ float Ps[8][16 * PSTR];   // 8.5 KB per-wave P scratch

  const int tid  = threadIdx.x;
  const int wave = tid >> 5;
  const int lane = tid & 31;
  const int ln   = lane & 15;
  const int hf   = lane >> 4;
  const int qb_per = nq_per_b / 128;
  const int b    = blockIdx.x / qb_per;
  const int qblk = blockIdx.x % qb_per;
  const int qrow = b * nq_per_b + qblk * 128 + wave * 16;
  const long kvbase0 = (long)b * nkv_per_b;
  const int  ntiles  = nkv_per_b >> 4;

  // Prologue: stage tile 0 into buffer 0 (overlaps with Q register loads)
  {
    const float* ksrc = Km + kvbase0 * NF;
    const float* vsrc = Vm + kvbase0 * NF;
    for (int g = tid; g < 512; g += 256) {   // 16 rows x 32 granules
      const int row = g >> 5, c4 = (g & 31) << 2;
      async_copy_b128(ksrc + row * NF + c4, &Ks[0][row * KSTR + c4]);
      async_copy_b128(vsrc + row * NF + c4, &Vs[0][row * VSTR + c4]);
    }
  }

  // Q slab for this wave, A-layout, in registers (64 VGPRs)
  v2f q[32];
  const float* qrowp = Q + (long)(qrow + ln) * NF;
#pragma unroll
  for (int c = 0; c < 32; ++c) {
    q[c].x = qrowp[4 * c + 2 * hf + 0];
    q[c].y = qrowp[4 * c + 2 * hf + 1];
  }

  v8f zero = {};
  v8f o[8];
#pragma unroll
  for (int ft = 0; ft < 8; ++ft) o[ft] = zero;
  float m[8], l[8];
#pragma unroll
  for (int r = 0; r < 8; ++r) { m[r] = -3.0e38f; l[r] = 0.0f; }

  for (int it = 0; it < ntiles; ++it) {
    const int buf = it & 1;
    async_wait0();     // my granules of tile `it` have landed
    __syncthreads();   // everyone's granules landed; prev iteration fully done

    // Issue DMA for tile it+1 into the other buffer (overlaps with compute)
    if (it + 1 < ntiles) {
      const float* ksrc = Km + (kvbase0 + (long)(it + 1) * 16) * NF;
      const float* vsrc = Vm + (kvbase0 + (long)(it + 1) * 16) * NF;
      for (int g = tid; g < 512; g += 256) {
        const int row = g >> 5, c4 = (g & 31) << 2;
        async_copy_b128(ksrc + row * NF + c4, &Ks[buf ^ 1][row * KSTR + c4]);
        async_copy_b128(vsrc + row * NF + c4, &Vs[buf ^ 1][row * VSTR + c4]);
      }
    }

    // S = Q @ K^T  (16x16 tile), accumulate over 128 channels
    v8f s = zero;
    const float* krow = &Ks[buf][ln * KSTR + 2 * hf];
#pragma unroll
    for (int c = 0; c < 32; ++c) {
      v2f bb;
      bb.x = krow[4 * c + 0];
      bb.y = krow[4 * c + 1];
      s = __builtin_amdgcn_wmma_f32_16x16x4_f32(
          false, q[c], false, bb, (short)0, s, false, false);
    }

    // Online softmax (rows r + 8*hf; N striped across the 16-lane group)
#pragma unroll
    for (int r = 0; r < 8; ++r) {
      float tmax = s[r];
      for (int off = 1; off < 16; off <<= 1)
        tmax = fmaxf(tmax, __shfl_xor(tmax, off, 16));
      const float mn    = fmaxf(m[r], tmax);
      const float scale = __expf(m[r] - mn);
      const float pv    = __expf(s[r] - mn);
      float rs = pv;
      for (int off = 1; off < 16; off <<= 1)
        rs += __shfl_xor(rs, off, 16);
      l[r] = l[r] * scale + rs;
      m[r] = mn;
#pragma unroll
      for (int ft = 0; ft < 8; ++ft) o[ft][r] *= scale;
      Ps[wave][(r + 8 * hf) * PSTR + ln] = pv;  // C-layout store of P
    }
    __syncthreads();  // conservative: order P store -> A-layout load

    // O += P @ V : A = P (16x16), B = V tile (16x128)
    v2f pa[4];
#pragma unroll
    for (int kc = 0; kc < 4; ++kc) {
      pa[kc].x = Ps[wave][ln * PSTR + 4 * kc + 2 * hf + 0];
      pa[kc].y = Ps[wave][ln * PSTR + 4 * kc + 2 * hf + 1];
    }
#pragma unroll
    for (int ft = 0; ft < 8; ++ft) {
#pragma unroll
      for (int kc = 0; kc < 4; ++kc) {
        v2f bb;
        bb.x = Vs[buf][(4 * kc + 2 * hf + 0) * VSTR + ft * 16 + ln];
        bb.y = Vs[buf][(4 * kc + 2 * hf + 1) * VSTR + ft * 16 + ln];
        o[ft] = __builtin_amdgcn_wmma_f32_16x16x4_f32(
            false, pa[kc], false, bb, (short)0, o[ft], false, false);
      }
    }
  }

  // Normalize by softmax denominator and store
#pragma unroll
  for (int ft = 0; ft < 8; ++ft)
#pragma unroll
    for (int r = 0; r < 8; ++r)
      O[(long)(qrow + r + 8 * hf) * NF + ft * 16 + ln] = o[ft][r] / l[r];
}

// ---------------------------------------------------------------------------
// Kernel 4: per-column mean/var over t[M,128] (population variance).
// ---------------------------------------------------------------------------
__global__ void colstats(const float* __restrict__ T,
                         float* __restrict__ stats,  // [0..127]=mean, [128..255]=var
                         int M) {
  const int col = blockIdx.x;
  float s = 0.f, s2 = 0.f;
  for (int r = threadIdx.x; r < M; r += 256) {
    const float v = T[(long)r * NF + col];
    s += v; s2 += v * v;
  }
  __shared__ float sh[256], sh2[256];
  sh[threadIdx.x] = s; sh2[threadIdx.x] = s2;
  __syncthreads();
  for (int ofs = 128; ofs > 0; ofs >>= 1) {
    if (threadIdx.x < ofs) {
      sh[threadIdx.x]  += sh[threadIdx.x + ofs];
      sh2[threadIdx.x] += sh2[threadIdx.x + ofs];
    }
    __syncthreads();
  }
  if (threadIdx.x == 0) {
    const float mean = sh[0] / (float)M;
    stats[col]       = mean;
    stats[NF + col]  = sh2[0] / (float)M - mean * mean;
  }
}

// ---------------------------------------------------------------------------
// Kernel 5: out = xdec + (t - mean)*rsqrt(var+eps)*gamma + beta
// ---------------------------------------------------------------------------
__global__ void bn_add(const float* __restrict__ T,
                       const float* __restrict__ xdec,
                       const float* __restrict__ stats,
                       const float* __restrict__ gamma,
                       const float* __restrict__ beta,
                       float* __restrict__ out, int M) {
  const long i = (long)blockIdx.x * blockDim.x + threadIdx.x;
  if (i >= (long)M * NF) return;
  const int col = (int)(i & (NF - 1));
  const float mean = stats[col];
  const float var  = stats[NF + col];
  out[i] = xdec[i] + (T[i] - mean) * rsqrtf(var + EPS) * gamma[col] + beta[col];
}

// ---------------------------------------------------------------------------
extern "C" void kernel_launch(void* const* d_in, const int* in_sizes, int n_in,
                              void* d_out, int out_size, void* d_ws, size_t ws_size,
                              hipStream_t stream) {
  const float* feat_dec = (const float*)d_in[0];  // [Nq,64]
  const float* feat_enc = (const float*)d_in[1];  // [Nkv,64]
  const int*   nbr      = (const int*)  d_in[2];  // [Nq,27]
  const float* W_p1     = (const float*)d_in[3];  // [27,64,128]
  const float* W_q      = (const float*)d_in[4];  // [128,128]
  const float* W_k      = (const float*)d_in[5];  // [64,128]
  const float* W_v      = (const float*)d_in[6];  // [64,128]
  const float* W_trans  = (const float*)d_in[7];  // [128,128]
  const float* gamma    = (const float*)d_in[8];  // [128]
  const float* beta     = (const float*)d_in[9];  // [128]

  const int Nq  = in_sizes[0] / 64;   // 16384
  const int Nkv = in_sizes[1] / 64;   // 32768
  const int B   = 8;

  float* ws    = (float*)d_ws;
  float* xdec  = ws;                   // Nq*128
  float* Qb    = xdec + (long)Nq * NF;
  float* Kb    = Qb   + (long)Nq * NF;
  float* Vb    = Kb   + (long)Nkv * NF;
  float* xr    = Vb   + (long)Nkv * NF;
  float* tbuf  = xr   + (long)Nq * NF;
  float* stats = tbuf + (long)Nq * NF; // 256 floats

  p1conv_wmma<<<Nq / 128, 256, 0, stream>>>(feat_dec, nbr, W_p1, xdec, Nq);
  gemm_wmma<<<Nq / 128, 256, 0, stream>>>(xdec, W_q, Qb, Nq, 128);
  gemm_wmma<<<Nkv / 128, 256, 0, stream>>>(feat_enc, W_k, Kb, Nkv, 64);
  gemm_wmma<<<Nkv / 128, 256, 0, stream>>>(feat_enc, W_v, Vb, Nkv, 64);
  attn_wmma<<<Nq / 128, 256, 0, stream>>>(Qb, Kb, Vb, xr, Nq / B, Nkv / B);
  gemm_wmma<<<Nq / 128, 256, 0, stream>>>(xr, W_trans, tbuf, Nq, 128);
  colstats<<<NF, 256, 0, stream>>>(tbuf, stats, Nq);
  bn_add<<<(int)(((long)Nq * NF + 255) / 256), 256, 0, stream>>>(
      tbuf, xdec, stats, gamma, beta, (float*)d_out, Nq);
}